// NeighborAwareConv_80599356277300
// MI455X (gfx1250) — compile-verified
//
#include <hip/hip_runtime.h>

// ---------------------------------------------------------------------------
// NeighborAwareConv for MI455X (gfx1250, wave32).
// Memory-bound scatter/gather: async global->LDS staging + f32 WMMA for the
// batched per-edge dot products; native f32 global atomics (forced via inline
// asm, device scope) for the segment sums.
// ---------------------------------------------------------------------------

typedef __attribute__((ext_vector_type(2))) float v2f;
typedef __attribute__((ext_vector_type(8))) float v8f;

#define D_DIM 128
#define WPB 2          // waves per block in the edge kernel
#define RS 132         // padded LDS row stride (floats): 16B-aligned, bank-safe

__device__ __forceinline__ unsigned lds_addr_of(const void* p) {
  // Generic pointers to LDS carry the workgroup-relative LDS byte offset in
  // their low 32 bits (flat->LDS mapping truncates to addr[31:0]).
  return (unsigned)(unsigned long long)p;
}

__device__ __forceinline__ void async_load_b128(unsigned lds_byte_addr,
                                                const void* gaddr) {
  unsigned long long ga = (unsigned long long)gaddr;
  asm volatile("global_load_async_to_lds_b128 %0, %1, off"
               :: "v"(lds_byte_addr), "v"(ga)
               : "memory");
}

__device__ __forceinline__ void wait_asynccnt0() {
  asm volatile("s_wait_asynccnt 0x0" ::: "memory");
}

// Non-returning device-scope f32 atomic add (single GLOBAL_ATOMIC_ADD_F32;
// avoids any CAS-loop fallback). RMW resolves at the device-coherent L2.
__device__ __forceinline__ void global_fadd(float* p, float v) {
  unsigned long long addr = (unsigned long long)p;
  asm volatile("global_atomic_add_f32 %0, %1, off scope:SCOPE_DEV"
               :: "v"(addr), "v"(v)
               : "memory");
}

// --------------------------- zero fill -------------------------------------
__global__ void zero_kernel(float* __restrict__ p, long long n) {
  long long i = (long long)blockIdx.x * blockDim.x + threadIdx.x;
  long long stride = (long long)gridDim.x * blockDim.x;
  for (; i < n; i += stride) p[i] = 0.0f;
}

// ------------------- per-row inverse L2 norm (wave per row) ----------------
__global__ void norm_kernel(const float* __restrict__ emb,
                            float* __restrict__ inv_norm, int nrows) {
  const int lane = threadIdx.x & 31;
  const int wid = threadIdx.x >> 5;
  const int row = blockIdx.x * (blockDim.x >> 5) + wid;
  if (row >= nrows) return;
  const float4 v = reinterpret_cast<const float4*>(emb + (size_t)row * D_DIM)[lane];
  float s = v.x * v.x + v.y * v.y + v.z * v.z + v.w * v.w;
#pragma unroll
  for (int off = 16; off > 0; off >>= 1) s += __shfl_xor(s, off);
  float r = rsqrtf(s);
  r = r * (1.5f - 0.5f * s * r * r);  // one Newton step
  if (lane == 0) inv_norm[row] = r;
}

// ------------- scatter normalized rows into S, count degrees --------------
__global__ void scatterS_kernel(const float* __restrict__ src_emb,
                                const float* __restrict__ inv_norm,
                                const int* __restrict__ src_idx,
                                const int* __restrict__ dst_idx,
                                float* __restrict__ S,
                                float* __restrict__ deg, int E) {
  const int lane = threadIdx.x & 31;
  const int wid = threadIdx.x >> 5;
  const int e = blockIdx.x * (blockDim.x >> 5) + wid;
  if (e >= E) return;
  const int s = src_idx[e];
  const int d = dst_idx[e];
  const float inv = inv_norm[s];
  const float4 v = reinterpret_cast<const float4*>(src_emb + (size_t)s * D_DIM)[lane];
  float* o = S + (size_t)d * D_DIM + lane * 4;
  global_fadd(o + 0, v.x * inv);
  global_fadd(o + 1, v.y * inv);
  global_fadd(o + 2, v.z * inv);
  global_fadd(o + 3, v.w * inv);
  if (lane == 0) global_fadd(deg + d, 1.0f);
}

// ---- per-edge cosine weight (WMMA f32 16x16x4) + weighted scatter --------
// One wave handles 16 edges: async-stage 16 src rows + 16 S rows into LDS,
// compute the 16 dots as the diagonal of a 16x16 WMMA product, then scatter
// w * src_row (reusing the staged LDS rows) into out via f32 atomics.
__global__ void edge_kernel(const float* __restrict__ src_emb,
                            const float* __restrict__ inv_norm,
                            const int* __restrict__ src_idx,
                            const int* __restrict__ dst_idx,
                            const float* __restrict__ S,
                            const float* __restrict__ deg,
                            float* __restrict__ out, int E) {
  __shared__ __align__(16) float sSrc[WPB][16 * RS];  // raw src rows
  __shared__ __align__(16) float sAgg[WPB][16 * RS];  // S[dst] rows
  __shared__ float sDot[WPB][16];
  __shared__ int sSi[WPB][16];
  __shared__ int sDi[WPB][16];

  const int lane = threadIdx.x & 31;
  const int wid = threadIdx.x >> 5;
  const int base = (blockIdx.x * WPB + wid) * 16;

  // edge metadata (clamped so all lanes issue valid loads; scatter is guarded)
  if (lane < 16) {
    int e = base + lane;
    if (e >= E) e = E - 1;
    sSi[wid][lane] = src_idx[e];
    sDi[wid][lane] = dst_idx[e];
  }
  __syncthreads();

  // Stage 16 src rows + 16 S rows: one async b128 per lane per row
  // (32 lanes x 16B = 512B row). Tracked by ASYNCcnt.
  for (int i = 0; i < 16; ++i) {
    const float* gs = src_emb + (size_t)sSi[wid][i] * D_DIM + lane * 4;
    const float* ga = S + (size_t)sDi[wid][i] * D_DIM + lane * 4;
    async_load_b128(lds_addr_of(&sSrc[wid][i * RS + lane * 4]), gs);
    async_load_b128(lds_addr_of(&sAgg[wid][i * RS + lane * 4]), ga);
  }
  wait_asynccnt0();
  __syncthreads();

  // WMMA f32 16x16x4: A lane layout -> lanes 0-15: row M=lane, K = k,k+1;
  // lanes 16-31: row M=lane-16, K = k+2,k+3. B mirrors A with N = lane%16.
  const int half = lane >> 4;
  const int m = lane & 15;
  const float invn = inv_norm[sSi[wid][m]];
  const float* arow = &sSrc[wid][m * RS];
  const float* brow = &sAgg[wid][m * RS];
  v8f acc = {};
#pragma unroll
  for (int k = 0; k < D_DIM; k += 4) {
    const int c0 = k + 2 * half;
    v2f a, b;
    a.x = arow[c0] * invn;
    a.y = arow[c0 + 1] * invn;
    b.x = brow[c0];
    b.y = brow[c0 + 1];
    acc = __builtin_amdgcn_wmma_f32_16x16x4_f32(
        /*neg_a=*/false, a, /*neg_b=*/false, b,
        /*c_mod=*/(short)0, acc, /*reuse_a=*/false, /*reuse_b=*/false);
  }

  // Extract diagonal of the 16x16 product:
  //   diag(i), i<8  -> VGPR i, lane i
  //   diag(8+r)     -> VGPR r, lane 24+r
#pragma unroll
  for (int r = 0; r < 8; ++r) {
    const float v = acc[r];
    if (lane == r) sDot[wid][r] = v;
    if (lane == r + 24) sDot[wid][r + 8] = v;
  }
  __syncthreads();

  // Weighted scatter: out[d] += relu(dot/degc) * src_row (raw, from LDS).
  for (int i = 0; i < 16; ++i) {
    const int e = base + i;
    if (e < E) {
      const int d = sDi[wid][i];
      const float degc = fmaxf(deg[d], 1.0f);
      const float w = fmaxf(sDot[wid][i] / degc, 0.0f);
      if (w != 0.0f) {  // wave-uniform branch
        const float4 v =
            *reinterpret_cast<const float4*>(&sSrc[wid][i * RS + lane * 4]);
        float* o = out + (size_t)d * D_DIM + lane * 4;
        global_fadd(o + 0, w * v.x);
        global_fadd(o + 1, w * v.y);
        global_fadd(o + 2, w * v.z);
        global_fadd(o + 3, w * v.w);
      }
    }
  }
}

// ----------------------- final deg^-0.5 scaling ----------------------------
__global__ void scale_kernel(float* __restrict__ out,
                             const float* __restrict__ deg, long long n) {
  long long i = (long long)blockIdx.x * blockDim.x + threadIdx.x;
  long long stride = (long long)gridDim.x * blockDim.x;
  for (; i < n; i += stride) {
    const float degc = fmaxf(deg[i >> 7], 1.0f);
    float r = rsqrtf(degc);
    r = r * (1.5f - 0.5f * degc * r * r);
    out[i] *= r;
  }
}

// ---------------------------------------------------------------------------
extern "C" void kernel_launch(void* const* d_in, const int* in_sizes, int n_in,
                              void* d_out, int out_size, void* d_ws,
                              size_t ws_size, hipStream_t stream) {
  (void)n_in; (void)out_size; (void)ws_size;
  const float* user_emb = (const float*)d_in[0];
  const float* item_emb = (const float*)d_in[1];
  const int* edge_user = (const int*)d_in[2];
  const int* edge_item = (const int*)d_in[3];
  const int N_U = in_sizes[0] / D_DIM;
  const int N_I = in_sizes[1] / D_DIM;
  const int E = in_sizes[2];

  float* out_user = (float*)d_out;                       // [N_U, 128]
  float* out_item = out_user + (size_t)N_U * D_DIM;      // [N_I, 128]

  float* ws = (float*)d_ws;
  float* inv_u = ws;                                     // N_U
  float* inv_i = inv_u + N_U;                            // N_I
  float* deg_u = inv_i + N_I;                            // N_U
  float* deg_i = deg_u + N_U;                            // N_I
  float* S_u = deg_i + N_I;                              // N_U*128
  float* S_i = S_u + (size_t)N_U * D_DIM;                // N_I*128

  // Zero accumulators + output (harness poisons once; we must re-zero).
  zero_kernel<<<2048, 256, 0, stream>>>(deg_u, N_U);
  zero_kernel<<<2048, 256, 0, stream>>>(deg_i, N_I);
  zero_kernel<<<2048, 256, 0, stream>>>(S_u, (long long)N_U * D_DIM);
  zero_kernel<<<2048, 256, 0, stream>>>(S_i, (long long)N_I * D_DIM);
  zero_kernel<<<2048, 256, 0, stream>>>((float*)d_out,
                                        (long long)(N_U + N_I) * D_DIM);

  // Inverse norms.
  norm_kernel<<<(N_U + 7) / 8, 256, 0, stream>>>(user_emb, inv_u, N_U);
  norm_kernel<<<(N_I + 7) / 8, 256, 0, stream>>>(item_emb, inv_i, N_I);

  // Segment sums of normalized rows + degrees (one wave per edge).
  scatterS_kernel<<<(E + 7) / 8, 256, 0, stream>>>(user_emb, inv_u, edge_user,
                                                   edge_item, S_i, deg_i, E);
  scatterS_kernel<<<(E + 7) / 8, 256, 0, stream>>>(item_emb, inv_i, edge_item,
                                                   edge_user, S_u, deg_u, E);

  // Edge weights (WMMA) + weighted scatter.
  const int waves = (E + 15) / 16;
  const int eblocks = (waves + WPB - 1) / WPB;
  edge_kernel<<<eblocks, WPB * 32, 0, stream>>>(user_emb, inv_u, edge_user,
                                                edge_item, S_i, deg_i,
                                                out_item, E);
  edge_kernel<<<eblocks, WPB * 32, 0, stream>>>(item_emb, inv_i, edge_item,
                                                edge_user, S_u, deg_u,
                                                out_user, E);

  // Final deg^-0.5 scaling.
  scale_kernel<<<2048, 256, 0, stream>>>(out_item, deg_i,
                                         (long long)N_I * D_DIM);
  scale_kernel<<<2048, 256, 0, stream>>>(out_user, deg_u,
                                         (long long)N_U * D_DIM);
}